// GPUNormalCalculationOperator_68186900791878
// MI455X (gfx1250) — compile-verified
//
#include <hip/hip_runtime.h>
#include <hip/hip_bf16.h>
#include <math.h>

#define EPSF 1e-8f
#define CHUNK 256
#define DW_PER_CHUNK (CHUNK * 3)   // 768 index dwords per chunk
#define DW_PER_WAVE  96            // 32 lanes * 3 dwords
#define GROUPS_PER_WAVE 24         // 96 dwords / 4-per-b128

// ---- CDNA5 async-to-LDS availability probe (compile-time) -------------------
#if defined(__has_builtin)
#if __has_builtin(__builtin_amdgcn_global_load_async_to_lds_b128)
#define HAVE_ASYNC_LDS 1
#endif
#endif
#ifndef HAVE_ASYNC_LDS
#define HAVE_ASYNC_LDS 0
#endif

#if HAVE_ASYNC_LDS
// Toolchain signature (probe-confirmed): (v4i32 AS1*, v4i32 AS3*, Imm, Imm)
typedef int v4i_t __attribute__((vector_size(16)));
typedef __attribute__((address_space(1))) v4i_t* gv4i_p;   // global
typedef __attribute__((address_space(3))) v4i_t* lv4i_p;   // LDS
#endif

// -----------------------------------------------------------------------------
// Kernel 1: zero the accumulator (d_out) with 128-bit stores.
// -----------------------------------------------------------------------------
__global__ __launch_bounds__(256) void k_zero(float* __restrict__ out, int n) {
    int i = blockIdx.x * blockDim.x + threadIdx.x;
    int n4 = n >> 2;
    if (i < n4) {
        float4 z; z.x = 0.f; z.y = 0.f; z.z = 0.f; z.w = 0.f;
        reinterpret_cast<float4*>(out)[i] = z;
    }
    int tail = n & 3;
    if (i < tail) out[(n4 << 2) + i] = 0.f;
}

// -----------------------------------------------------------------------------
// Kernel 2: per-triangle face normal + scatter-add via L2 atomics.
// Index stream staged via gfx1250 async DMA (GLOBAL_LOAD_ASYNC_TO_LDS_B128),
// double-buffered and wave-private: no __syncthreads, overlap via ASYNCcnt.
// -----------------------------------------------------------------------------
__global__ __launch_bounds__(256) void k_face_scatter(const float* __restrict__ verts,
                                                      const int*   __restrict__ idx,
                                                      float*       __restrict__ accum,
                                                      int M) {
#if HAVE_ASYNC_LDS
    __shared__ int s_idx[2][DW_PER_CHUNK];
#endif
    const int tid = threadIdx.x;
    const int chunk_stride = gridDim.x * CHUNK;

#if HAVE_ASYNC_LDS
    const int wv   = tid >> 5;            // wave id within block (wave32)
    const int lane = tid & 31;
    const int woff = wv * DW_PER_WAVE + lane * 4;  // this lane's staging dword
    int buf = 0;

    // Prologue: stage first chunk into buffer 0 (wave-private region).
    {
        const int b = blockIdx.x * CHUNK;
        if (b < M) {
            int avail  = (M - b) * 3;
            int staged = ((avail < DW_PER_CHUNK) ? avail : DW_PER_CHUNK) & ~3;
            if (lane < GROUPS_PER_WAVE && woff + 4 <= staged) {
                __builtin_amdgcn_global_load_async_to_lds_b128(
                    (gv4i_p)(idx + (size_t)b * 3 + woff),
                    (lv4i_p)(&s_idx[0][woff]), 0, 0);
            }
        }
    }
#endif

    for (int base = blockIdx.x * CHUNK; base < M; base += chunk_stride) {
        const int t = base + tid;

        // L2 prefetch two chunks ahead (k+1 is covered by the async DMA).
        {
            int pb = base + 2 * chunk_stride;
            if (pb + tid < M)
                __builtin_prefetch(&idx[(size_t)(pb + tid) * 3], 0, 0);
        }

        int i0 = 0, i1 = 0, i2 = 0;

#if HAVE_ASYNC_LDS
        {
            int avail_cur  = (M - base) * 3;
            const int staged_cur =
                ((avail_cur < DW_PER_CHUNK) ? avail_cur : DW_PER_CHUNK) & ~3;

            // Issue DMA for chunk k+1 into the other buffer, then wait so that
            // chunk k (this wave's older async load) has landed; k+1 stays
            // in flight (in-order per-wave completion => asynccnt<=1 suffices).
            bool wave_issued_next = false;
            {
                const int nb = base + chunk_stride;
                if (nb < M) {
                    int avail  = (M - nb) * 3;
                    int staged = ((avail < DW_PER_CHUNK) ? avail : DW_PER_CHUNK) & ~3;
                    if (lane < GROUPS_PER_WAVE && woff + 4 <= staged) {
                        __builtin_amdgcn_global_load_async_to_lds_b128(
                            (gv4i_p)(idx + (size_t)nb * 3 + woff),
                            (lv4i_p)(&s_idx[buf ^ 1][woff]), 0, 0);
                    }
                    wave_issued_next = (wv * DW_PER_WAVE + 4) <= staged;  // uniform
                }
            }
#if defined(__has_builtin) && __has_builtin(__builtin_amdgcn_s_wait_asynccnt)
            if (wave_issued_next) __builtin_amdgcn_s_wait_asynccnt(1);
            else                  __builtin_amdgcn_s_wait_asynccnt(0);
#else
            if (wave_issued_next) asm volatile("s_wait_asynccnt 1" ::: "memory");
            else                  asm volatile("s_wait_asynccnt 0" ::: "memory");
#endif
            if (t < M) {
                if (tid * 3 + 3 <= staged_cur) {   // staged coverage is monotone
                    i0 = s_idx[buf][tid * 3 + 0];
                    i1 = s_idx[buf][tid * 3 + 1];
                    i2 = s_idx[buf][tid * 3 + 2];
                } else {                            // ragged tail of last chunk
                    i0 = idx[(size_t)t * 3 + 0];
                    i1 = idx[(size_t)t * 3 + 1];
                    i2 = idx[(size_t)t * 3 + 2];
                }
            }
            buf ^= 1;
        }
#else
        if (t < M) {
            i0 = idx[(size_t)t * 3 + 0];
            i1 = idx[(size_t)t * 3 + 1];
            i2 = idx[(size_t)t * 3 + 2];
        }
#endif

        if (t < M) {
            // Random gathers: 12 B per vertex; vertex set (24 MB) is L2-resident.
            const float ax = verts[(size_t)i0 * 3 + 0];
            const float ay = verts[(size_t)i0 * 3 + 1];
            const float az = verts[(size_t)i0 * 3 + 2];
            const float bx = verts[(size_t)i1 * 3 + 0];
            const float by = verts[(size_t)i1 * 3 + 1];
            const float bz = verts[(size_t)i1 * 3 + 2];
            const float cx = verts[(size_t)i2 * 3 + 0];
            const float cy = verts[(size_t)i2 * 3 + 1];
            const float cz = verts[(size_t)i2 * 3 + 2];

            const float e1x = bx - ax, e1y = by - ay, e1z = bz - az;
            const float e2x = cx - ax, e2y = cy - ay, e2z = cz - az;

            float nx = e1y * e2z - e1z * e2y;
            float ny = e1z * e2x - e1x * e2z;
            float nz = e1x * e2y - e1y * e2x;

            const float len = sqrtf(nx * nx + ny * ny + nz * nz);
            const float inv = 1.0f / fmaxf(len, EPSF);
            nx *= inv; ny *= inv; nz *= inv;

            // Non-returning f32 atomics -> GLOBAL_ATOMIC_ADD_F32 (L2 RMW units).
            atomicAdd(&accum[(size_t)i0 * 3 + 0], nx);
            atomicAdd(&accum[(size_t)i0 * 3 + 1], ny);
            atomicAdd(&accum[(size_t)i0 * 3 + 2], nz);
            atomicAdd(&accum[(size_t)i1 * 3 + 0], nx);
            atomicAdd(&accum[(size_t)i1 * 3 + 1], ny);
            atomicAdd(&accum[(size_t)i1 * 3 + 2], nz);
            atomicAdd(&accum[(size_t)i2 * 3 + 0], nx);
            atomicAdd(&accum[(size_t)i2 * 3 + 1], ny);
            atomicAdd(&accum[(size_t)i2 * 3 + 2], nz);
        }
    }
}

// -----------------------------------------------------------------------------
// Kernel 3: in-place per-vertex normalization of the accumulator.
// -----------------------------------------------------------------------------
__global__ __launch_bounds__(256) void k_normalize(float* __restrict__ nrm, int N) {
    int v = blockIdx.x * blockDim.x + threadIdx.x;
    if (v >= N) return;
    const size_t o = (size_t)v * 3;
    const float x = nrm[o + 0];
    const float y = nrm[o + 1];
    const float z = nrm[o + 2];
    const float inv = 1.0f / fmaxf(sqrtf(x * x + y * y + z * z), EPSF);
    nrm[o + 0] = x * inv;
    nrm[o + 1] = y * inv;
    nrm[o + 2] = z * inv;
}

// -----------------------------------------------------------------------------
extern "C" void kernel_launch(void* const* d_in, const int* in_sizes, int n_in,
                              void* d_out, int out_size, void* d_ws, size_t ws_size,
                              hipStream_t stream) {
    (void)d_ws; (void)ws_size; (void)n_in;

    const float* verts = (const float*)d_in[0];
    const int*   idx   = (const int*)d_in[1];
    float*       out   = (float*)d_out;

    const int N = in_sizes[0] / 3;   // vertex count
    const int M = in_sizes[1] / 3;   // triangle count

    // 1) zero accumulator (d_out)
    {
        int n4 = out_size >> 2;
        int blocks = (n4 + 255) / 256;
        if (blocks < 1) blocks = 1;
        k_zero<<<blocks, 256, 0, stream>>>(out, out_size);
    }

    // 2) face-normal scatter: 4096 blocks -> ~4 chunk iterations each, so the
    //    double-buffered DMA pipeline has real loop trips; 32K waves in flight.
    {
        int chunks = (M + CHUNK - 1) / CHUNK;
        int blocks = chunks < 4096 ? chunks : 4096;
        if (blocks < 1) blocks = 1;
        k_face_scatter<<<blocks, CHUNK, 0, stream>>>(verts, idx, out, M);
    }

    // 3) normalize per vertex, in place
    {
        int blocks = (N + 255) / 256;
        if (blocks < 1) blocks = 1;
        k_normalize<<<blocks, 256, 0, stream>>>(out, N);
    }
}